// ConditionalVQEmbedding_51196010169093
// MI455X (gfx1250) — compile-verified
//
#include <hip/hip_runtime.h>

// Problem constants (from the reference)
#define NB     32          // batch
#define DD     256         // feature dim
#define NN     4096        // H*W positions per batch
#define KK     512         // codes per condition
#define NCOND  16
#define TILE_N 64          // positions per block
#define BLOCK  128         // 4 wave32 waves
#define BPAD   260         // LDS row pad: stride%64 == 4 -> conflict-free b64 reads

typedef __attribute__((ext_vector_type(2))) float v2f;
typedef __attribute__((ext_vector_type(8))) float v8f;

// ---------------------------------------------------------------------------
// Kernel 0: halfnorm[c][k] = 0.5 * sum_d weight[c][k][d]^2
// ---------------------------------------------------------------------------
__global__ __launch_bounds__(KK) void vq_halfnorm_kernel(const float* __restrict__ w,
                                                         float* __restrict__ hn) {
    const int c = blockIdx.x;        // condition
    const int k = threadIdx.x;       // 0..511
    const float* row = w + ((size_t)c * KK + k) * DD;
    float s = 0.0f;
#pragma unroll 8
    for (int d = 0; d < DD; ++d) {
        float v = row[d];
        s += v * v;
    }
    hn[c * KK + k] = 0.5f * s;
}

// ---------------------------------------------------------------------------
// Kernel 1: distances via fp32 WMMA, argmin over K, gather + double write-out
//   scores[k][n] = 0.5*||e_k||^2 - dot(z_n, e_k)   (argmin-equivalent)
// ---------------------------------------------------------------------------
__global__ __launch_bounds__(BLOCK) void vq_main_kernel(const float* __restrict__ z,
                                                        const int*   __restrict__ C,
                                                        const float* __restrict__ w,
                                                        const float* __restrict__ hn,
                                                        float*       __restrict__ out) {
    __shared__ float zt[TILE_N][BPAD];   // z tile, transposed to [n_local][d]
    __shared__ float cbt[16][BPAD];      // current 16-code tile [k_local][d]
    __shared__ float hn_s[16];
    __shared__ int   idx_s[TILE_N];

    const int tid  = threadIdx.x;
    const int bb   = blockIdx.y;                 // batch
    const int n0   = blockIdx.x * TILE_N;        // first position of this tile
    const int cond = C[bb];
    const size_t cb_base = (size_t)cond * KK * DD;
    const float* zb = z + (size_t)bb * DD * NN;  // z[b][d][n]

    // ---- stage z tile (global is d-major; transpose into LDS, n-coalesced) ----
    for (int i = tid; i < DD * TILE_N; i += BLOCK) {
        const int d = i >> 6;               // / TILE_N
        const int j = i & (TILE_N - 1);
        zt[j][d] = zb[(size_t)d * NN + n0 + j];
    }

    const int lane = tid & 31;
    const int wave = tid >> 5;       // 0..3 -> 16 positions each
    const int l15  = lane & 15;
    const int hi   = lane >> 4;      // lanes 16..31 carry K+2 (A/B) and M+8 (C/D)
    const int nl   = wave * 16 + l15;

    float best  = 3.402823466e+38f;
    int   bestk = 0;

    for (int kt = 0; kt < KK / 16; ++kt) {
        __syncthreads();   // zt ready (iter 0) / previous tile's WMMAs done
        // ---- stage 16-code tile + its half-norms ----
        for (int i = tid; i < 16 * DD; i += BLOCK) {
            const int r = i >> 8;           // / DD
            const int c = i & (DD - 1);
            cbt[r][c] = w[cb_base + (size_t)(kt * 16 + r) * DD + c];
        }
        if (tid < 16) hn_s[tid] = hn[cond * KK + kt * 16 + tid];
        __syncthreads();

        // software prefetch of next code tile -> global_prefetch_b8
        if (tid < 16 && (kt + 1) < KK / 16)
            __builtin_prefetch(&w[cb_base + (size_t)((kt + 1) * 16 + tid) * DD], 0, 1);

        // ---- fp32 WMMA: A = 16(k) x 4(d) codes, B = 4(d) x 16(n) z ----
        v8f acc = {};
        const float* arow = &cbt[l15][0];
        const float* brow = &zt[nl][0];
#pragma unroll 8
        for (int d0 = 0; d0 < DD; d0 += 4) {
            const v2f a = *(const v2f*)(arow + d0 + 2 * hi);  // K=d0+2hi, d0+2hi+1
            const v2f b = *(const v2f*)(brow + d0 + 2 * hi);
            acc = __builtin_amdgcn_wmma_f32_16x16x4_f32(
                /*neg_a=*/false, a, /*neg_b=*/false, b,
                /*c_mod=*/(short)0, acc, /*reuse_a=*/false, /*reuse_b=*/false);
        }

        // ---- per-lane running argmin over this tile's 8 candidate codes ----
#pragma unroll
        for (int v = 0; v < 8; ++v) {
            const float score = hn_s[hi * 8 + v] - acc[v];
            const int   k     = kt * 16 + hi * 8 + v;
            if (score < best) { best = score; bestk = k; }
        }
    }

    // ---- combine lane pair (l, l^16): both hold candidates for the same n ----
    {
        const float obest = __shfl_xor(best, 16, 32);
        const int   okk   = __shfl_xor(bestk, 16, 32);
        if (obest < best || (obest == best && okk < bestk)) { best = obest; bestk = okk; }
    }
    if (hi == 0) idx_s[nl] = bestk;
    __syncthreads();

    // ---- gather winning rows, write both outputs (forward values identical) ----
    float* out2 = out + (size_t)NB * DD * NN;
    for (int i = tid; i < DD * TILE_N; i += BLOCK) {
        const int d = i >> 6;
        const int j = i & (TILE_N - 1);
        const float val = w[cb_base + (size_t)idx_s[j] * DD + d];
        const size_t o = ((size_t)bb * DD + d) * NN + n0 + j;
        out[o]  = val;   // z_q_x
        out2[o] = val;   // z_q_x_bar
    }
}

// ---------------------------------------------------------------------------
extern "C" void kernel_launch(void* const* d_in, const int* in_sizes, int n_in,
                              void* d_out, int out_size, void* d_ws, size_t ws_size,
                              hipStream_t stream) {
    const float* z  = (const float*)d_in[0];   // z_e_x  [32,256,64,64] f32
    const int*   C  = (const int*)d_in[1];     // C      [32] int
    const float* w  = (const float*)d_in[2];   // weight [16,512,256] f32
    float*       out = (float*)d_out;          // 2 x [32,256,64,64] f32
    float*       hn  = (float*)d_ws;           // 16*512 f32 half-norms

    vq_halfnorm_kernel<<<dim3(NCOND), dim3(KK), 0, stream>>>(w, hn);
    vq_main_kernel<<<dim3(NN / TILE_N, NB), dim3(BLOCK), 0, stream>>>(z, C, w, hn, out);
}